// AbstractEGCN_26577257628252
// MI455X (gfx1250) — compile-verified
//
#include <hip/hip_runtime.h>
#include <hip/hip_bf16.h>

typedef __attribute__((ext_vector_type(2))) float v2f;
typedef __attribute__((ext_vector_type(4))) float v4f;
typedef __attribute__((ext_vector_type(8))) float v8f;

#define HIDDEN 64
#define OUTF   64

// ---------------------------------------------------------------------------
// Phase 1: Y[n, 0:64]  = x[n] . W[:, 0:64]^T   (i.e. Wcat row n,   n <  64)
//          Y[n,64:128] = x[n] . W[:,64:128]^T  (i.e. Wcat row n-64, n >= 64)
// Y is [n_nodes, 128] f32 in workspace. One wave computes one 16x16 tile of Y
// using V_WMMA_F32_16X16X4_F32 (f32 A 16x4, f32 B 4x16, f32 C/D 16x16).
//
// f32 A-fragment lane layout (ISA 7.12.2): lane l holds A[m=l&15, kb] in .x and
// A[m, kb+1] in .y with kb = (l>>4)*2.  B-fragment: lane l holds B[kb, n=l&15]
// and B[kb+1, n].  C/D: VGPR v, lanes 0-15 -> M=v, lanes 16-31 -> M=v+8.
// ---------------------------------------------------------------------------
__global__ void __launch_bounds__(256)
node_gemm_kernel(const float* __restrict__ x,    // [n_nodes, 64]
                 const float* __restrict__ W,    // [64, 128]
                 float* __restrict__ Y,          // [n_nodes, 128]
                 int n_nodes)
{
    const int lane  = threadIdx.x & 31;
    const int wave  = threadIdx.x >> 5;
    const int tile  = blockIdx.x * 8 + wave;     // grid sized exactly: tiles % 8 == 0
    const int m_tile = tile >> 3;                // 16-row tile of Y
    const int n_tile = tile & 7;                 // 16-col tile of Y (128 cols total)

    const int half  = lane >> 4;                 // 0 or 1
    const int idx16 = lane & 15;
    const int kb    = half * 2;                  // K sub-offset {0,2}

    // A rows come from x (clamped so every lane has a valid address; EXEC stays all-1)
    int m  = m_tile * 16 + idx16;
    int mc = m < n_nodes ? m : (n_nodes - 1);
    const float* __restrict__ arow = x + (size_t)mc * HIDDEN;

    // B[k][n] = Wcat[n][k]; Wcat[n][k] = W[(n&63)*128 + (n>=64 ? 64+k : k)]
    const int n = n_tile * 16 + idx16;           // 0..127
    const float* __restrict__ brow = W + (size_t)(n & 63) * 128 + ((n >= 64) ? 64 : 0);

    v8f acc = {};
#pragma unroll
    for (int k = 0; k < HIDDEN; k += 4) {
        v2f a = *(const v2f*)(arow + k + kb);
        v2f b = *(const v2f*)(brow + k + kb);
        acc = __builtin_amdgcn_wmma_f32_16x16x4_f32(
            /*neg_a=*/false, a, /*neg_b=*/false, b,
            /*c_mod=*/(short)0, acc, /*reuse_a=*/false, /*reuse_b=*/false);
    }

    // Scatter C: VGPR v -> row (m_tile*16 + v + half*8), col n_tile*16 + idx16
    const int col = n_tile * 16 + idx16;
#pragma unroll
    for (int v = 0; v < 8; ++v) {
        int row = m_tile * 16 + v + half * 8;
        if (row < n_nodes)
            Y[(size_t)row * 128 + col] = acc[v];
    }
}

// ---------------------------------------------------------------------------
// Phase 2: out[e, :] = Y[src[e], 0:64] + Y[dst[e], 64:128] + b
// 16 lanes per edge, one float4-worth per lane. Y gathers hit L2 (51 MB < 192 MB).
// Output stream (410 MB) uses non-temporal stores to avoid polluting L2.
// ---------------------------------------------------------------------------
__global__ void __launch_bounds__(256)
edge_combine_kernel(const float* __restrict__ Y,
                    const int* __restrict__ ei,   // [2, E]
                    const float* __restrict__ bias,
                    float* __restrict__ out,      // [E, 64]
                    int E)
{
    long long t = (long long)blockIdx.x * blockDim.x + threadIdx.x;
    long long e = t >> 4;
    int q = (int)(t & 15);                        // 128-bit slot 0..15
    if (e >= E) return;

    int s = ei[e];
    int d = ei[(size_t)E + e];

    v4f a  = ((const v4f*)(Y + (size_t)s * 128))[q];
    v4f c  = ((const v4f*)(Y + (size_t)d * 128 + 64))[q];
    v4f vb = ((const v4f*)bias)[q];

    v4f r = a + c + vb;

    __builtin_nontemporal_store(r, (v4f*)(out + (size_t)e * 64) + q);
}

// ---------------------------------------------------------------------------
// Fallback: fused gather + WMMA GEMM over the concatenated K=128 dimension,
// used only if the workspace can't hold Y. One wave per 16-edge tile,
// 4 accumulators (full 64 outputs), 128 WMMA issues per tile (32 k-steps x 4).
// ---------------------------------------------------------------------------
__global__ void __launch_bounds__(256)
fused_edge_gemm_kernel(const float* __restrict__ x,
                       const int* __restrict__ ei,
                       const float* __restrict__ W,
                       const float* __restrict__ bias,
                       float* __restrict__ out,
                       int E)
{
    const int lane = threadIdx.x & 31;
    const int wave = threadIdx.x >> 5;
    const long long tile = (long long)blockIdx.x * 8 + wave;
    const long long e0 = tile * 16;
    if (e0 >= E) return;                          // wave-uniform exit

    const int half  = lane >> 4;
    const int idx16 = lane & 15;
    const int kb    = half * 2;

    long long em = e0 + idx16;
    if (em > E - 1) em = E - 1;                   // clamp, keep EXEC all-1
    const int s = ei[em];
    const int d = ei[(size_t)E + em];
    const float* __restrict__ srow = x + (size_t)s * HIDDEN;
    const float* __restrict__ drow = x + (size_t)d * HIDDEN;

    v8f acc0 = {}, acc1 = {}, acc2 = {}, acc3 = {};
#pragma unroll 4
    for (int k = 0; k < 128; k += 4) {
        // kb in {0,2}, k % 4 == 0 -> (k+kb) never straddles the 64 boundary
        const float* ap = (k + kb) < 64 ? (srow + k + kb) : (drow + k + kb - 64);
        v2f a = *(const v2f*)ap;

        v2f b0 = *(const v2f*)(W + (size_t)(0 * 16 + idx16) * 128 + k + kb);
        v2f b1 = *(const v2f*)(W + (size_t)(1 * 16 + idx16) * 128 + k + kb);
        v2f b2 = *(const v2f*)(W + (size_t)(2 * 16 + idx16) * 128 + k + kb);
        v2f b3 = *(const v2f*)(W + (size_t)(3 * 16 + idx16) * 128 + k + kb);

        acc0 = __builtin_amdgcn_wmma_f32_16x16x4_f32(false, a, false, b0, (short)0, acc0, false, false);
        acc1 = __builtin_amdgcn_wmma_f32_16x16x4_f32(false, a, false, b1, (short)0, acc1, false, false);
        acc2 = __builtin_amdgcn_wmma_f32_16x16x4_f32(false, a, false, b2, (short)0, acc2, false, false);
        acc3 = __builtin_amdgcn_wmma_f32_16x16x4_f32(false, a, false, b3, (short)0, acc3, false, false);
    }

#pragma unroll
    for (int v = 0; v < 8; ++v) {
        long long erow = e0 + v + half * 8;
        if (erow < E) {
            float* orow = out + (size_t)erow * OUTF;
            orow[ 0 + idx16] = acc0[v] + bias[ 0 + idx16];
            orow[16 + idx16] = acc1[v] + bias[16 + idx16];
            orow[32 + idx16] = acc2[v] + bias[32 + idx16];
            orow[48 + idx16] = acc3[v] + bias[48 + idx16];
        }
    }
}

extern "C" void kernel_launch(void* const* d_in, const int* in_sizes, int n_in,
                              void* d_out, int out_size, void* d_ws, size_t ws_size,
                              hipStream_t stream) {
    const float* x    = (const float*)d_in[0];   // [n_nodes, 64]
    const int*   ei   = (const int*)d_in[1];     // [2, E]
    const float* W    = (const float*)d_in[2];   // [64, 128]
    const float* bias = (const float*)d_in[3];   // [64]
    float* out = (float*)d_out;

    const int n_nodes = in_sizes[0] / HIDDEN;
    const int E       = in_sizes[1] / 2;

    const size_t need = (size_t)n_nodes * 128 * sizeof(float);
    if (ws_size >= need) {
        float* Y = (float*)d_ws;

        // Phase 1: node GEMM -> Y.  m_tiles * 8 n_tiles, 8 waves (tiles) per block.
        const int m_tiles = (n_nodes + 15) / 16;
        const int tiles   = m_tiles * 8;                   // always divisible by 8
        node_gemm_kernel<<<tiles / 8, 256, 0, stream>>>(x, W, Y, n_nodes);

        // Phase 2: per-edge gather-add, 16 lanes per edge.
        const long long total = (long long)E * 16;
        const int blocks = (int)((total + 255) / 256);
        edge_combine_kernel<<<blocks, 256, 0, stream>>>(Y, ei, bias, out, E);
    } else {
        // Fallback: fused gather + WMMA (no workspace needed).
        const long long e_tiles = ((long long)E + 15) / 16;
        const int blocks = (int)((e_tiles + 7) / 8);
        fused_edge_gemm_kernel<<<blocks, 256, 0, stream>>>(x, ei, W, bias, out, E);
    }
}